// CSRSparseRetrievalModelIterativeThreadSafe_57947698758425
// MI455X (gfx1250) — compile-verified
//
#include <hip/hip_runtime.h>

#define TOPK   10
#define TPB    256
#define CHUNK  1024   // posting-list elements staged per block via async->LDS
#define B1     512    // phase-1 top-k blocks

typedef unsigned long long u64;
typedef unsigned int       u32;

// ---------- monotone float<->u32 key mapping (order-preserving, incl. negatives)
__device__ __forceinline__ u32 mapf(float v) {
  u32 b = __float_as_uint(v);
  return (b & 0x80000000u) ? ~b : (b | 0x80000000u);
}
__device__ __forceinline__ float unmapf(u32 m) {
  u32 b = (m & 0x80000000u) ? (m & 0x7FFFFFFFu) : ~m;
  return __uint_as_float(b);
}

// key = (mapped_val << 32) | ~doc  -> max key = max val, tie -> smaller doc id
__device__ __forceinline__ u64 make_key(float v, int idx) {
  return ((u64)mapf(v) << 32) | (u32)(~(u32)idx);
}

// register-resident sorted (desc) top-10 insert: 1 cmp in the common case
__device__ __forceinline__ void ins10(u64 k,
    u64& t0, u64& t1, u64& t2, u64& t3, u64& t4,
    u64& t5, u64& t6, u64& t7, u64& t8, u64& t9) {
  if (k <= t9) return;
  t9 = k; u64 tmp;
  if (t9 > t8) { tmp = t8; t8 = t9; t9 = tmp; }
  if (t8 > t7) { tmp = t7; t7 = t8; t8 = tmp; }
  if (t7 > t6) { tmp = t6; t6 = t7; t7 = tmp; }
  if (t6 > t5) { tmp = t5; t5 = t6; t6 = tmp; }
  if (t5 > t4) { tmp = t4; t4 = t5; t5 = tmp; }
  if (t4 > t3) { tmp = t3; t3 = t4; t4 = tmp; }
  if (t3 > t2) { tmp = t2; t2 = t3; t3 = tmp; }
  if (t2 > t1) { tmp = t1; t1 = t2; t2 = tmp; }
  if (t1 > t0) { tmp = t0; t0 = t1; t1 = tmp; }
}

// ------------------------------------------------- zero the acc (b128 stores)
__global__ void zero_acc_kernel(float* __restrict__ acc, const int* __restrict__ ndp) {
  const int n   = *ndp;
  const int nv  = n >> 2;                       // float4 count
  const int gid = blockIdx.x * blockDim.x + threadIdx.x;
  const int str = gridDim.x * blockDim.x;
  float4* acc4 = (float4*)acc;
  const float4 z = make_float4(0.f, 0.f, 0.f, 0.f);
  for (int i = gid; i < nv; i += str) acc4[i] = z;
  for (int i = (nv << 2) + gid; i < n; i += str) acc[i] = 0.0f;  // tail (<4)
}

// ------------------------------------------------- scatter: postings -> acc
// blockIdx.y = query slot q, blockIdx.x = chunk of that term's posting list.
// Stage CHUNK (doc,val) entries into LDS with async global->LDS b128 loads,
// then issue HW f32 atomics into the L2-resident accumulator.
__global__ void scatter_kernel(const int*   __restrict__ ccol,
                               const int*   __restrict__ rind,
                               const float* __restrict__ cval,
                               const int*   __restrict__ qidx,
                               const float* __restrict__ qval,
                               float*       __restrict__ acc,
                               long long nnz) {
  __shared__ int   sdoc[CHUNK];
  __shared__ float sval[CHUNK];

  const int   q     = blockIdx.y;
  const int   term  = qidx[q];
  const float w     = qval[q];
  const int   start = ccol[term];
  const int   len   = ccol[term + 1] - start;
  const int   cbase = blockIdx.x * CHUNK;
  const int   tid   = threadIdx.x;

  // each lane async-loads one 16B granule (4 elements) of each array into LDS
  const int  g   = cbase + tid * 4;                       // granule start in list
  const bool gok = ((long long)start + g + 4 <= nnz);     // stays inside the array
  const u32  goff = (u32)((long long)(start + (gok ? g : 0)) * 4);
  const u32  ldd  = (u32)(size_t)(&sdoc[tid * 4]);        // low 32 bits = LDS offset
  const u32  ldv  = (u32)(size_t)(&sval[tid * 4]);

  asm volatile("global_load_async_to_lds_b128 %0, %1, %2"
               :: "v"(ldd), "v"(goff), "s"(rind) : "memory");
  asm volatile("global_load_async_to_lds_b128 %0, %1, %2"
               :: "v"(ldv), "v"(goff), "s"(cval) : "memory");
  asm volatile("s_wait_asynccnt 0x0" ::: "memory");
  __syncthreads();

#pragma unroll
  for (int j = 0; j < CHUNK / TPB; ++j) {
    const int e  = tid + j * TPB;        // element within chunk (LDS conflict-free)
    const int le = cbase + e;            // element within posting list
    if (le >= len) continue;
    const int  gran = (e >> 2) << 2;     // granule that loaded element e
    const bool ok   = ((long long)start + cbase + gran + 4 <= nnz);
    int doc; float v;
    if (ok) { doc = sdoc[e]; v = sval[e]; }
    else {   // rare array-edge granule: fall back to direct load
      const long long gi = (long long)start + le;
      doc = rind[gi]; v = cval[gi];
    }
    __hip_atomic_fetch_add(&acc[doc], v * w,
                           __ATOMIC_RELAXED, __HIP_MEMORY_SCOPE_AGENT);
  }
}

// ------------------------------- block-level merge of per-thread top-10 lists
__device__ __forceinline__ void block_merge_top10(
    u64 t0, u64 t1, u64 t2, u64 t3, u64 t4,
    u64 t5, u64 t6, u64 t7, u64 t8, u64 t9,
    u64* __restrict__ out_keys) {          // 10 keys written by thread 0
  __shared__ u64 cand[TPB * TOPK];
  __shared__ u64 red[TPB];
  __shared__ int rslot[TPB];
  const int tid = threadIdx.x;
  const int b   = tid * TOPK;
  cand[b + 0] = t0; cand[b + 1] = t1; cand[b + 2] = t2; cand[b + 3] = t3;
  cand[b + 4] = t4; cand[b + 5] = t5; cand[b + 6] = t6; cand[b + 7] = t7;
  cand[b + 8] = t8; cand[b + 9] = t9;
  __syncthreads();
  for (int r = 0; r < TOPK; ++r) {
    u64 best = 0; int slot = b;
#pragma unroll
    for (int j = 0; j < TOPK; ++j) {
      u64 c = cand[b + j];
      if (c > best) { best = c; slot = b + j; }
    }
    red[tid] = best; rslot[tid] = slot;
    __syncthreads();
    for (int s = TPB / 2; s > 0; s >>= 1) {
      if (tid < s && red[tid + s] > red[tid]) {
        red[tid] = red[tid + s]; rslot[tid] = rslot[tid + s];
      }
      __syncthreads();
    }
    if (tid == 0) { out_keys[r] = red[0]; cand[rslot[0]] = 0; }
    __syncthreads();
  }
}

// ----------------------------------------- phase 1: acc -> per-block top-10
// b128 (float4) streaming scan of the accumulator; tail handled scalar.
__global__ void topk_phase1(const float* __restrict__ acc,
                            const int*   __restrict__ ndp,
                            u64*         __restrict__ cands) {
  const int n   = *ndp;
  const int nv  = n >> 2;
  const int gid = blockIdx.x * blockDim.x + threadIdx.x;
  const int str = gridDim.x * blockDim.x;
  u64 t0=0,t1=0,t2=0,t3=0,t4=0,t5=0,t6=0,t7=0,t8=0,t9=0;
  const float4* acc4 = (const float4*)acc;
  for (int i = gid; i < nv; i += str) {
    const float4 v = acc4[i];
    const int base = i << 2;
    ins10(make_key(v.x, base + 0), t0,t1,t2,t3,t4,t5,t6,t7,t8,t9);
    ins10(make_key(v.y, base + 1), t0,t1,t2,t3,t4,t5,t6,t7,t8,t9);
    ins10(make_key(v.z, base + 2), t0,t1,t2,t3,t4,t5,t6,t7,t8,t9);
    ins10(make_key(v.w, base + 3), t0,t1,t2,t3,t4,t5,t6,t7,t8,t9);
  }
  for (int i = (nv << 2) + gid; i < n; i += str)   // tail (<4 elements)
    ins10(make_key(acc[i], i), t0,t1,t2,t3,t4,t5,t6,t7,t8,t9);
  block_merge_top10(t0,t1,t2,t3,t4,t5,t6,t7,t8,t9, cands + (size_t)blockIdx.x * TOPK);
}

// -------------------------- phase 2: candidates -> final top-10 (vals + idx)
__global__ void topk_phase2(const u64* __restrict__ cands, int ncand,
                            float* __restrict__ out) {
  u64 t0=0,t1=0,t2=0,t3=0,t4=0,t5=0,t6=0,t7=0,t8=0,t9=0;
  for (int i = threadIdx.x; i < ncand; i += blockDim.x)
    ins10(cands[i], t0,t1,t2,t3,t4,t5,t6,t7,t8,t9);
  __shared__ u64 final_keys[TOPK];
  block_merge_top10(t0,t1,t2,t3,t4,t5,t6,t7,t8,t9, final_keys);
  if (threadIdx.x == 0) {
    for (int r = 0; r < TOPK; ++r) {
      const u64 k = final_keys[r];
      out[r]        = unmapf((u32)(k >> 32));
      out[TOPK + r] = (float)(~(u32)k);   // doc id < 2^24 -> exact in f32
    }
  }
}

extern "C" void kernel_launch(void* const* d_in, const int* in_sizes, int n_in,
                              void* d_out, int out_size, void* d_ws, size_t ws_size,
                              hipStream_t stream) {
  const int*   ccol = (const int*)  d_in[0];
  const int*   rind = (const int*)  d_in[1];
  const float* cval = (const float*)d_in[2];
  const int*   qidx = (const int*)  d_in[3];
  const float* qval = (const float*)d_in[4];
  const int*   ndp  = (const int*)  d_in[5];   // n_docs (device scalar)

  const int       nterms = in_sizes[0] - 1;
  const long long nnz    = (long long)in_sizes[1];
  const int       Q      = in_sizes[3];
  const int       L      = (int)(nnz / nterms);  // uniform posting-list length

  float* acc = (float*)d_ws;                                   // n_docs floats
  const size_t cand_bytes = (size_t)B1 * TOPK * sizeof(u64);   // 40 KB at ws end
  const size_t cand_off   = (ws_size - cand_bytes) & ~(size_t)7;
  u64* cands = (u64*)((char*)d_ws + cand_off);

  zero_acc_kernel<<<1024, TPB, 0, stream>>>(acc, ndp);

  const int chunks = (L + CHUNK - 1) / CHUNK;
  scatter_kernel<<<dim3(chunks, Q), TPB, 0, stream>>>(ccol, rind, cval,
                                                      qidx, qval, acc, nnz);

  topk_phase1<<<B1, TPB, 0, stream>>>(acc, ndp, cands);
  topk_phase2<<<1,  TPB, 0, stream>>>(cands, B1 * TOPK, (float*)d_out);
}